// Column_20298015441325
// MI455X (gfx1250) — compile-verified
//
#include <hip/hip_runtime.h>
#include <hip/hip_bf16.h>

// ---------------------------------------------------------------------------
// Problem constants (from the reference)
// ---------------------------------------------------------------------------
#define T_DIM   32
#define K_DIM   512
#define D_DIM   100000          // 1 * 50 * 2000
#define THRESH  10.0f
#define KWTA    16

// Split-K GEMM decomposition
#define CHUNK   800             // D elements per workgroup (multiple of 80)
#define DCHUNKS (D_DIM / CHUNK) // 125
#define KTILES  (K_DIM / 16)    // 32

typedef float v2f __attribute__((ext_vector_type(2)));
typedef float v8f __attribute__((ext_vector_type(8)));

// ---------------------------------------------------------------------------
// Zero the potential accumulator (split-K partial sums land here via atomics)
// ---------------------------------------------------------------------------
__global__ void Column_zero_ws(float* __restrict__ p, int n) {
    int i = blockIdx.x * blockDim.x + threadIdx.x;
    if (i < n) p[i] = 0.0f;
}

// ---------------------------------------------------------------------------
// GEMM: pot[t,k] = sum_d A[t,d] * W[k,d]   via V_WMMA_F32_16X16X4_F32
//
// One wave per block. blockIdx.x selects a 16-wide k tile, blockIdx.y an
// 800-long slice of the 100k reduction dim. Two 16x16 f32 accumulators cover
// t=0..15 and t=16..31.
//
// fp32 WMMA operand layout (wave32):
//   A (16x4): lanes 0-15 -> M=lane, hold K={0,1}; lanes 16-31 -> M=lane-16,
//             hold K={2,3}.   B (4x16) mirrors it with N across lanes.
// => every lane loads an aligned float2 at (row_base + d + 2*(lane>>4)).
//
// All inner-loop loads use pointer + compile-time immediate offsets (fits the
// 24-bit signed IOFFSET, incl. the t+16 row at +6,400,000 B) so the steady
// state is pure global_load_b64 + v_wmma with no per-iteration address VALU.
// W is loaded non-temporally: it streams once (204.8 MB) while rec_field
// (12.8 MB, reused by 32 k-tiles) stays resident in the 192 MB L2.
// ---------------------------------------------------------------------------
__global__ __launch_bounds__(32)
void Column_gemm_tk(const float* __restrict__ A,   // rec_field  (T, D)
                    const float* __restrict__ W,   // weight     (K, D)
                    float* __restrict__ pot) {     // accumulator (T, K)
    const int lane = threadIdx.x;      // 0..31
    const int half = lane >> 4;        // 0 | 1
    const int m    = lane & 15;        // 0..15
    const int co   = 2 * half;         // K-offset inside a 4-wide WMMA step

    const int k0 = blockIdx.x * 16;
    const int d0 = blockIdx.y * CHUNK;

    // Per-lane stream bases (advanced by +80 floats per outer iteration).
    const float* pA = A + (size_t)m * D_DIM + d0 + co;           // t = m
    const float* pW = W + (size_t)(k0 + m) * D_DIM + d0 + co;    // k = k0+m

    v8f acc0 = {0.f, 0.f, 0.f, 0.f, 0.f, 0.f, 0.f, 0.f};
    v8f acc1 = acc0;

    for (int it = 0; it < CHUNK / 80; ++it) {
        // stream-ahead hint on the weight rows (~1.3 KB ahead)
        __builtin_prefetch(pW + 320, 0, 1);
#pragma unroll
        for (int u = 0; u < 80; u += 4) {
            v2f a0 = *(const v2f*)(pA + u);                      // imm offset
            v2f a1 = *(const v2f*)(pA + u + 16 * D_DIM);         // imm 6.4e6 B
            v2f b  = __builtin_nontemporal_load((const v2f*)(pW + u)); // NT
            // (neg_a, A, neg_b, B, c_mod, C, reuse_a, reuse_b)
            acc0 = __builtin_amdgcn_wmma_f32_16x16x4_f32(
                       false, a0, false, b, (short)0, acc0, false, false);
            acc1 = __builtin_amdgcn_wmma_f32_16x16x4_f32(
                       false, a1, false, b, (short)0, acc1, false, false);
        }
        pA += 80;
        pW += 80;
    }

    // C/D layout: lane gives N = m, rows: VGPR r -> M = r + 8*half
    const int k = k0 + m;
#pragma unroll
    for (int r = 0; r < 8; ++r) {
        const int t0 = r + 8 * half;
        atomicAdd(&pot[t0 * K_DIM + k],        acc0[r]);
        atomicAdd(&pot[(t0 + 16) * K_DIM + k], acc1[r]);
    }
}

// ---------------------------------------------------------------------------
// Finalize: threshold -> nspk/first/values -> v -> total -> 16 argmax rounds
// -> coef -> out = sign(pot * coef).  One block of 512 threads (k = tid).
// ---------------------------------------------------------------------------
__global__ __launch_bounds__(K_DIM)
void Column_finalize(const float* __restrict__ pot,  // (T, K) raw projections
                     float* __restrict__ out) {      // (T, K) output
    __shared__ float s_tot[K_DIM];
    __shared__ float s_coef[K_DIM];
    __shared__ float s_val[K_DIM];
    __shared__ int   s_idx[K_DIM];

    const int k = threadIdx.x;

    // nspk = number of timesteps with potential > THRESH
    int nspk = 0;
    for (int t = 0; t < T_DIM; ++t) {
        if (pot[t * K_DIM + k] > THRESH) ++nspk;
    }
    // first = clip(T - nspk, 0, T-1); values = thresholded pot at 'first'
    int first = T_DIM - nspk;
    if (first > T_DIM - 1) first = T_DIM - 1;
    if (first < 0) first = 0;
    float of     = pot[first * K_DIM + k];
    float values = (of > THRESH) ? of : 0.0f;

    // v = max_k(values * sign(nspk)) * T
    s_val[k] = (nspk > 0) ? values : 0.0f;
    __syncthreads();
    for (int s = K_DIM / 2; s > 0; s >>= 1) {
        if (k < s) s_val[k] = fmaxf(s_val[k], s_val[k + s]);
        __syncthreads();
    }
    const float v = s_val[0] * (float)T_DIM;
    __syncthreads();

    s_tot[k]  = (float)nspk * values + (float)nspk * v;
    s_coef[k] = 0.0f;
    __syncthreads();

    // KWTA sequential rounds; argmax = first occurrence of max (ties -> min idx)
    for (int round = 0; round < KWTA; ++round) {
        s_val[k] = s_tot[k];
        s_idx[k] = k;
        __syncthreads();
        for (int s = K_DIM / 2; s > 0; s >>= 1) {
            if (k < s) {
                float v1 = s_val[k], v2 = s_val[k + s];
                int   i1 = s_idx[k], i2 = s_idx[k + s];
                if (v2 > v1 || (v2 == v1 && i2 < i1)) {
                    s_val[k] = v2;
                    s_idx[k] = i2;
                }
            }
            __syncthreads();
        }
        if (k == 0) {
            int i = s_idx[0];
            if (s_tot[i] != 0.0f) {   // valid winner
                s_tot[i]  = 0.0f;     // suppress feature
                s_coef[i] = 1.0f;
            }
        }
        __syncthreads();
    }

    // out[t,k] = sign(pot_thresholded * coef) in {0,1}
    const float c = s_coef[k];
    for (int t = 0; t < T_DIM; ++t) {
        float o = pot[t * K_DIM + k];
        out[t * K_DIM + k] = (o > THRESH && c > 0.0f) ? 1.0f : 0.0f;
    }
}

// ---------------------------------------------------------------------------
// Launch
// ---------------------------------------------------------------------------
extern "C" void kernel_launch(void* const* d_in, const int* in_sizes, int n_in,
                              void* d_out, int out_size, void* d_ws, size_t ws_size,
                              hipStream_t stream) {
    const float* rec_field = (const float*)d_in[0];  // (32, 1, 50, 2000)
    const float* weight    = (const float*)d_in[1];  // (512, 1, 50, 2000)
    float*       out       = (float*)d_out;          // (32, 512, 1, 1)
    float*       pot_ws    = (float*)d_ws;           // T*K fp32 accumulator

    const int potN = T_DIM * K_DIM;  // 16384

    Column_zero_ws<<<(potN + 255) / 256, 256, 0, stream>>>(pot_ws, potN);

    dim3 grid(KTILES, DCHUNKS);      // 32 x 125 single-wave workgroups
    Column_gemm_tk<<<grid, 32, 0, stream>>>(rec_field, weight, pot_ws);

    Column_finalize<<<1, K_DIM, 0, stream>>>(pot_ws, out);
}